// VQVAE2_25503515804109
// MI455X (gfx1250) — compile-verified
//
#include <hip/hip_runtime.h>
#include <hip/hip_bf16.h>
#include <math.h>

typedef __attribute__((ext_vector_type(2))) float v2f;
typedef __attribute__((ext_vector_type(8))) float v8f;

__device__ __forceinline__ v8f wmma_f32_16x16x4(v2f a, v2f b, v8f c) {
  // (neg_a, A, neg_b, B, c_mod, C, reuse_a, reuse_b) -> v_wmma_f32_16x16x4_f32
  return __builtin_amdgcn_wmma_f32_16x16x4_f32(false, a, false, b, (short)0, c, false, false);
}

#define V8F_ZERO {0.f,0.f,0.f,0.f,0.f,0.f,0.f,0.f}

// ---------------------------------------------------------------------------
// Conv1d (stride S, kernel width KW<=4, pad 1) as implicit GEMM on WMMA f32.
// Each wave computes TWO 16x16 output tiles (32 out-channels x 16 positions)
// sharing one B fragment: 2x arithmetic intensity, two independent WMMA
// accumulation chains. K step = 4 = one input channel x 4 taps.
// ---------------------------------------------------------------------------
template<int CI, int CO, int LI, int LO, int STRIDE, int KW, bool RELU>
__global__ __launch_bounds__(256) void conv_wmma_k(
    const float* __restrict__ in, const float* __restrict__ w,
    const float* __restrict__ bias, float* __restrict__ out, int B)
{
  constexpr int OP = CO / 32;               // o-tile PAIRS
  constexpr int LT = LO / 16;
  const int wid  = (int)((blockIdx.x * blockDim.x + threadIdx.x) >> 5);
  const int lane = (int)(threadIdx.x & 31);
  const int total = B * OP * LT;
  if (wid >= total) return;                 // wave-uniform: EXEC all-ones inside
  const int lt = wid % LT;
  const int op = (wid / LT) % OP;
  const int b  = wid / (LT * OP);
  const int lr = lane & 15, half = lane >> 4;
  const int l  = lt * 16 + lr;              // output position (N col)
  const int t0 = 2 * half;                  // K-pair base tap
  const int px = STRIDE * l - 1 + t0;       // input position of pair
  const bool vx = (px >= 0) && (px < LI);
  const bool vy = (px + 1 < LI);
  const float* __restrict__ inb = in + (size_t)b * CI * LI + px;
  const float* __restrict__ wb0 = w + (size_t)(op * 32 + lr) * CI * KW + t0;
  const float* __restrict__ wb1 = wb0 + (size_t)16 * CI * KW;
  v8f acc0 = V8F_ZERO, acc1 = V8F_ZERO;
#pragma unroll 8
  for (int i = 0; i < CI; ++i) {
    const float* ip = inb + (size_t)i * LI;
    v2f bm;
    bm.x = vx ? ip[0] : 0.f;
    bm.y = vy ? ip[1] : 0.f;
    v2f a0, a1;
    a0.x = wb0[(size_t)i * KW];                                 // t0 < KW always
    a0.y = (t0 + 1 < KW) ? wb0[(size_t)i * KW + 1] : 0.f;       // zero-pad KW=3
    a1.x = wb1[(size_t)i * KW];
    a1.y = (t0 + 1 < KW) ? wb1[(size_t)i * KW + 1] : 0.f;
    acc0 = wmma_f32_16x16x4(a0, bm, acc0);
    acc1 = wmma_f32_16x16x4(a1, bm, acc1);
  }
  float* __restrict__ ob = out + (size_t)b * CO * LO + (size_t)lt * 16 + lr;
#pragma unroll
  for (int v = 0; v < 8; ++v) {
    const int oc0 = op * 32 + v + 8 * half; // C layout: M = v (+8 high half)
    const int oc1 = oc0 + 16;
    float r0 = acc0[v] + bias[oc0];
    float r1 = acc1[v] + bias[oc1];
    if (RELU) { r0 = fmaxf(r0, 0.f); r1 = fmaxf(r1, 0.f); }
    ob[(size_t)oc0 * LO] = r0;
    ob[(size_t)oc1 * LO] = r1;
  }
}

// ---------------------------------------------------------------------------
// ConvTranspose1d (k=4, s=2, p=1): out[l] gets 2 taps. Even l: (j=l/2-1,t=3),
// (j=l/2,t=1); odd l: (j=(l-1)/2,t=2),(j+1,t=0). Waves handle 16 same-parity
// outputs (uniform weight taps, contiguous input pair) and TWO o-tiles.
// K = 2*CI, k-step = 4 = two channels x 2 taps (one channel per lane-half).
// ---------------------------------------------------------------------------
template<int CI, int CO, int LIN, bool RELU>
__global__ __launch_bounds__(256) void deconv_wmma_k(
    const float* __restrict__ in, const float* __restrict__ w, /* [CI,CO,4] */
    const float* __restrict__ bias, float* __restrict__ out, int B)
{
  constexpr int LOUT = LIN * 2;
  constexpr int OP = CO / 32;    // o-tile pairs
  constexpr int PT = LIN / 16;   // 16 same-parity outputs per tile
  constexpr int KS = CI / 2;     // k-steps of 4 over K=2*CI
  const int wid  = (int)((blockIdx.x * blockDim.x + threadIdx.x) >> 5);
  const int lane = (int)(threadIdx.x & 31);
  const int total = B * OP * 2 * PT;
  if (wid >= total) return;
  const int pt  = wid % PT;
  const int par = (wid / PT) & 1;
  const int op  = (wid / (PT * 2)) % OP;
  const int b   = wid / (PT * 2 * OP);
  const int lr = lane & 15, half = lane >> 4;
  const int l  = (pt * 16 + lr) * 2 + par;  // output position (N col)
  const int o0 = op * 32 + lr;              // first out-channel row (A)
  const int j0 = (l >> 1) - 1 + par;        // first input tap position
  const bool v0 = (j0 >= 0);
  const bool v1 = (j0 + 1 < LIN);
  const int tA = 3 - par, tB = 1 - par;     // weight taps for (j0, j0+1)
  const float* __restrict__ inb = in + (size_t)b * CI * LIN + j0;
  v8f acc0 = V8F_ZERO, acc1 = V8F_ZERO;
#pragma unroll 8
  for (int kk = 0; kk < KS; ++kk) {
    const int i = 2 * kk + half;            // channel for this lane-half
    const float* ip = inb + (size_t)i * LIN;
    v2f bm;
    bm.x = v0 ? ip[0] : 0.f;
    bm.y = v1 ? ip[1] : 0.f;
    const float* wp0 = w + ((size_t)i * CO + o0) * 4;
    const float* wp1 = wp0 + 16 * 4;
    v2f a0, a1;
    a0.x = wp0[tA]; a0.y = wp0[tB];
    a1.x = wp1[tA]; a1.y = wp1[tB];
    acc0 = wmma_f32_16x16x4(a0, bm, acc0);
    acc1 = wmma_f32_16x16x4(a1, bm, acc1);
  }
  float* __restrict__ ob = out + (size_t)b * CO * LOUT + l;
#pragma unroll
  for (int v = 0; v < 8; ++v) {
    const int oc0 = op * 32 + v + 8 * half;
    const int oc1 = oc0 + 16;
    float r0 = acc0[v] + bias[oc0];
    float r1 = acc1[v] + bias[oc1];
    if (RELU) { r0 = fmaxf(r0, 0.f); r1 = fmaxf(r1, 0.f); }
    ob[(size_t)oc0 * LOUT] = r0;
    ob[(size_t)oc1 * LOUT] = r1;
  }
}

// ---------------------------------------------------------------------------
// Precompute ||emb_k||^2.
// ---------------------------------------------------------------------------
__global__ void emb_norm_k(const float* __restrict__ emb, float* __restrict__ en) {
  const int k = (int)(blockIdx.x * blockDim.x + threadIdx.x);
  if (k < 512) {
    float s = 0.f;
    const float* e = emb + (size_t)k * 64;
#pragma unroll 8
    for (int d = 0; d < 64; ++d) s += e[d] * e[d];
    en[k] = s;
  }
}

// ---------------------------------------------------------------------------
// VQ argmin: one wave per 32 rows (two A-fragment sets sharing each emb B
// fragment -> halves emb traffic, two independent WMMA chains). Distance
// GEMM [32,64]x[64,512] in WMMA f32 16x16x4; score = ||e||^2 - 2*dot.
// Running first-min per lane, then 16-lane shfl_xor tree per half.
// ---------------------------------------------------------------------------
__global__ __launch_bounds__(256) void quant_argmin_k(
    const float* __restrict__ z, const float* __restrict__ emb,
    const float* __restrict__ en, int* __restrict__ idx, int N)
{
  const int wid  = (int)((blockIdx.x * blockDim.x + threadIdx.x) >> 5);
  const int lane = (int)(threadIdx.x & 31);
  const int rowBase = wid * 32;
  if (rowBase >= N) return;
  const int lr = lane & 15, half = lane >> 4;
  // A fragments: row = rowBase+lr (+16), K pair base = 4*kk + 2*half (8B aligned)
  v2f afr0[16], afr1[16];
  const float* zr0 = z + (size_t)(rowBase + lr) * 64 + 2 * half;
  const float* zr1 = zr0 + 16 * 64;
#pragma unroll
  for (int kk = 0; kk < 16; ++kk) {
    afr0[kk] = *(const v2f*)(zr0 + 4 * kk);
    afr1[kk] = *(const v2f*)(zr1 + 4 * kk);
  }

  float bV0[8], bV1[8]; int bI0[8], bI1[8];
#pragma unroll
  for (int v = 0; v < 8; ++v) {
    bV0[v] = 3.4e38f; bI0[v] = 0;
    bV1[v] = 3.4e38f; bI1[v] = 0;
  }

  for (int ct = 0; ct < 32; ++ct) {          // 32 tiles of 16 codes
    v8f acc0 = V8F_ZERO, acc1 = V8F_ZERO;
    const float* er = emb + (size_t)(ct * 16 + lr) * 64 + 2 * half;
#pragma unroll
    for (int kk = 0; kk < 16; ++kk) {
      v2f bfr = *(const v2f*)(er + 4 * kk);
      acc0 = wmma_f32_16x16x4(afr0[kk], bfr, acc0);
      acc1 = wmma_f32_16x16x4(afr1[kk], bfr, acc1);
    }
    const int code = ct * 16 + lr;           // C col = lane&15
    const float e = en[code];
#pragma unroll
    for (int v = 0; v < 8; ++v) {
      const float s0 = e - 2.f * acc0[v];
      const float s1 = e - 2.f * acc1[v];
      if (s0 < bV0[v]) { bV0[v] = s0; bI0[v] = code; }  // first-min kept
      if (s1 < bV1[v]) { bV1[v] = s1; bI1[v] = code; }
    }
  }
  // reduce over the 16 lanes holding the same rows (halves are independent)
#pragma unroll
  for (int m = 8; m >= 1; m >>= 1) {
#pragma unroll
    for (int v = 0; v < 8; ++v) {
      float ov; int oi;
      ov = __shfl_xor(bV0[v], m, 32); oi = __shfl_xor(bI0[v], m, 32);
      if (ov < bV0[v] || (ov == bV0[v] && oi < bI0[v])) { bV0[v] = ov; bI0[v] = oi; }
      ov = __shfl_xor(bV1[v], m, 32); oi = __shfl_xor(bI1[v], m, 32);
      if (ov < bV1[v] || (ov == bV1[v] && oi < bI1[v])) { bV1[v] = ov; bI1[v] = oi; }
    }
  }
  if (lr == 0) {
#pragma unroll
    for (int v = 0; v < 8; ++v) {
      idx[rowBase + v + 8 * half]      = bI0[v];
      idx[rowBase + 16 + v + 8 * half] = bI1[v];
    }
  }
}

// ---------------------------------------------------------------------------
// q = emb[idx]; accumulate sum((q-z)^2) for the VQ loss (x1.25/numel later).
// ---------------------------------------------------------------------------
__global__ __launch_bounds__(256) void gather_loss_k(
    const float* __restrict__ z, const float* __restrict__ emb,
    const int* __restrict__ idx, float* __restrict__ q,
    float* __restrict__ lossAcc, int N)
{
  const int n = (int)(blockIdx.x * blockDim.x + threadIdx.x);
  float s = 0.f;
  if (n < N) {
    const int k = idx[n];
    const float* e  = emb + (size_t)k * 64;
    const float* zr = z + (size_t)n * 64;
    float* qr = q + (size_t)n * 64;
#pragma unroll 8
    for (int d = 0; d < 64; ++d) {
      const float qq = e[d];
      qr[d] = qq;
      const float df = qq - zr[d];
      s += df * df;
    }
  }
  __shared__ float red[256];
  red[threadIdx.x] = s;
  __syncthreads();
  for (int st = 128; st > 0; st >>= 1) {
    if ((int)threadIdx.x < st) red[threadIdx.x] += red[threadIdx.x + st];
    __syncthreads();
  }
  if (threadIdx.x == 0) atomicAdd(lossAcc, red[0]);
}

// ---------------------------------------------------------------------------
// Final ConvTranspose1d 64->1 + tanh (tiny: 67 MMAC, bandwidth bound).
// ---------------------------------------------------------------------------
__global__ __launch_bounds__(256) void dec3_tanh_k(
    const float* __restrict__ in /*[B,64,4096]*/, const float* __restrict__ w /*[64,1,4]*/,
    const float* __restrict__ bias, float* __restrict__ out, int B)
{
  const int t = (int)(blockIdx.x * blockDim.x + threadIdx.x);
  if (t >= B * 8192) return;
  const int b = t >> 13;
  const int l = t & 8191;
  const int par = l & 1;
  const int j0 = (l >> 1) - 1 + par;
  const int tA = 3 - par, tB = 1 - par;
  float s = bias[0];
  const float* ib = in + (size_t)b * 64 * 4096;
#pragma unroll 8
  for (int i = 0; i < 64; ++i) {
    const float x0 = (j0 >= 0)       ? ib[(size_t)i * 4096 + j0]     : 0.f;
    const float x1 = (j0 + 1 < 4096) ? ib[(size_t)i * 4096 + j0 + 1] : 0.f;
    s += x0 * w[i * 4 + tA] + x1 * w[i * 4 + tB];
  }
  out[t] = tanhf(s);
}

__global__ void finalize_k(const float* __restrict__ acc, float* __restrict__ out, float scale) {
  if (threadIdx.x == 0) {
    out[0] = acc[0] * scale;
    out[1] = acc[1] * scale;
  }
}

// ---------------------------------------------------------------------------
extern "C" void kernel_launch(void* const* d_in, const int* in_sizes, int n_in,
                              void* d_out, int out_size, void* d_ws, size_t ws_size,
                              hipStream_t stream) {
  (void)in_sizes; (void)n_in; (void)out_size; (void)ws_size;
  const float* x    = (const float*)d_in[0];
  const float* e1w1 = (const float*)d_in[1];
  const float* e1b1 = (const float*)d_in[2];
  const float* e1w2 = (const float*)d_in[3];
  const float* e1b2 = (const float*)d_in[4];
  const float* e1w3 = (const float*)d_in[5];
  const float* e1b3 = (const float*)d_in[6];
  const float* e2w  = (const float*)d_in[7];
  const float* e2b  = (const float*)d_in[8];
  const float* emb1 = (const float*)d_in[9];
  const float* emb2 = (const float*)d_in[10];
  const float* dw1  = (const float*)d_in[11];
  const float* db1  = (const float*)d_in[12];
  const float* dw2  = (const float*)d_in[13];
  const float* db2  = (const float*)d_in[14];
  const float* dw3  = (const float*)d_in[15];
  const float* db3  = (const float*)d_in[16];
  float* out = (float*)d_out;

  // workspace layout (floats); bufA: h1 then d2, bufB: h2 then d1
  float* ws   = (float*)d_ws;
  float* bufA = ws;                  // 16,777,216
  float* bufB = ws + 16777216;       // 16,777,216
  float* ze1  = ws + 33554432;       //  4,194,304
  float* zq1  = ws + 37748736;
  float* ze2  = ws + 41943040;
  float* zq2  = ws + 46137344;
  float* en1  = ws + 50331648;       // 512
  float* en2  = en1 + 512;           // 512
  float* lossAcc = en2 + 512;        // 2
  int* idx1 = (int*)(lossAcc + 2);   // 65536
  int* idx2 = idx1 + 65536;          // 65536

  const int B = 64;
  auto blocks = [](long waves) { return (int)((waves * 32 + 255) / 256); };

  hipMemsetAsync(lossAcc, 0, 2 * sizeof(float), stream);
  emb_norm_k<<<2, 256, 0, stream>>>(emb1, en1);
  emb_norm_k<<<2, 256, 0, stream>>>(emb2, en2);

  // encoder1 (waves = B * (CO/32) * (LO/16))
  conv_wmma_k<1,  64, 8192, 4096, 2, 4, true ><<<blocks((long)B*2*256), 256, 0, stream>>>(x,    e1w1, e1b1, bufA, B);
  conv_wmma_k<64, 128, 4096, 2048, 2, 4, true ><<<blocks((long)B*4*128), 256, 0, stream>>>(bufA, e1w2, e1b2, bufB, B);
  conv_wmma_k<128, 64, 2048, 1024, 2, 4, false><<<blocks((long)B*2*64),  256, 0, stream>>>(bufB, e1w3, e1b3, ze1,  B);

  // VQ 1 (waves = N/32)
  quant_argmin_k<<<blocks(65536 / 32), 256, 0, stream>>>(ze1, emb1, en1, idx1, 65536);
  gather_loss_k<<<65536 / 256, 256, 0, stream>>>(ze1, emb1, idx1, zq1, lossAcc + 0, 65536);

  // encoder2 (k=3 conv zero-padded to KW=4 K-structure)
  conv_wmma_k<64, 64, 1024, 1024, 1, 3, true><<<blocks((long)B*2*64), 256, 0, stream>>>(zq1, e2w, e2b, ze2, B);

  // VQ 2
  quant_argmin_k<<<blocks(65536 / 32), 256, 0, stream>>>(ze2, emb2, en2, idx2, 65536);
  gather_loss_k<<<65536 / 256, 256, 0, stream>>>(ze2, emb2, idx2, zq2, lossAcc + 1, 65536);

  // decoder (waves = B * (CO/32) * 2 * (LIN/16))
  deconv_wmma_k<64, 128, 1024, true><<<blocks((long)B*4*2*64),  256, 0, stream>>>(zq2,  dw1, db1, bufB, B);
  deconv_wmma_k<128, 64, 2048, true><<<blocks((long)B*2*2*128), 256, 0, stream>>>(bufB, dw2, db2, bufA, B);
  dec3_tanh_k<<<(B * 8192) / 256, 256, 0, stream>>>(bufA, dw3, db3, out, B);

  // losses: 1.25 * mean((q-z)^2), numel = 64*64*1024
  finalize_k<<<1, 32, 0, stream>>>(lossAcc, out + 524288, 1.25f / 4194304.f);
}